// SpliceosomeModel_30666066494039
// MI455X (gfx1250) — compile-verified
//
#include <hip/hip_runtime.h>
#include <hip/hip_bf16.h>

// Problem dimensions (fixed by the reference)
#define GG 128   // genes
#define SS 4096  // splice sites per gene
#define CC 128   // channels
#define JJ 2048  // junctions per gene
#define TT 64    // transcripts per gene
#define KK 32    // junctions per transcript
#define HH 128   // hidden dim
#define C2 256   // 2*C (donor||acceptor concat)

typedef float v2f __attribute__((ext_vector_type(2)));
typedef float v8f __attribute__((ext_vector_type(8)));

// ---------------------------------------------------------------------------
// Kernel 1: gather junction embeddings + Linear(256->128)+ReLU+Linear(128->1)
// One wave (32 lanes) computes a 16-junction strip via v_wmma_f32_16x16x4_f32.
// 8 waves / block -> 128 junctions per block, W1^T staged in 128 KB LDS.
// ---------------------------------------------------------------------------
__global__ __launch_bounds__(256, 1)
void splice_junction_mlp(const float* __restrict__ reps,     // [G,S,C]
                         const float* __restrict__ gstart,   // [C]
                         const float* __restrict__ gend,     // [C]
                         const float* __restrict__ W1,       // [2C,H] row-major
                         const float* __restrict__ b1,       // [H]
                         const float* __restrict__ W2,       // [H] (H x 1)
                         const float* __restrict__ b2,       // [1]
                         const int*   __restrict__ don_idx,  // [G,J]
                         const int*   __restrict__ acc_idx,  // [G,J]
                         float*       __restrict__ pot)      // [G,J] out
{
    // W1 transposed: w1t[n][k] = W1[k][n]  -> contiguous-K float2 LDS loads for B frags
    __shared__ float w1t[HH][C2];                 // 128*256*4 = 128 KB
    for (int i = threadIdx.x; i < C2 * HH; i += 256) {
        int k = i >> 7;      // 0..255 (row of W1, coalesced global read)
        int n = i & 127;     // 0..127
        w1t[n][k] = W1[i];
    }
    __syncthreads();

    const int wave = threadIdx.x >> 5;            // wave32
    const int lane = threadIdx.x & 31;
    const int row  = lane & 15;                   // M row (A) / N col (B,C)
    const int half = lane >> 4;                   // K-split half per ISA layout
    const int g    = blockIdx.x >> 4;             // JJ/128 = 16 tiles per gene
    const int tile = blockIdx.x & 15;
    const int j0   = tile * 128 + wave * 16;      // first junction of this strip

    // Per-lane gather row pointers (lanes l and l+16 share the same junction,
    // they split the K dimension: half 0 -> K%4 in {0,1}, half 1 -> {2,3}).
    const int j  = j0 + row;
    const int dj = don_idx[g * JJ + j];
    const int aj = acc_idx[g * JJ + j];
    const float* drow = (dj < SS) ? (reps + ((size_t)g * SS + dj) * CC)
                                  : (dj == SS ? gstart : gend);
    const float* arow = (aj < SS) ? (reps + ((size_t)g * SS + aj) * CC)
                                  : (aj == SS ? gstart : gend);
    const int klane = half * 2;

    // C/D accumulators: acc[nt] covers N = nt*16 + (lane%16), M = v + 8*half.
    v8f acc[8];
#pragma unroll
    for (int nt = 0; nt < 8; ++nt) {
        float bv = b1[nt * 16 + row];             // bias depends on column N only
        acc[nt] = (v8f){bv, bv, bv, bv, bv, bv, bv, bv};
    }

    // K = 0..127  : donor channels
#pragma unroll 2
    for (int k0 = 0; k0 < CC; k0 += 4) {
        v2f a = *(const v2f*)(drow + k0 + klane); // A frag: A[row][klane..klane+1]
#pragma unroll
        for (int nt = 0; nt < 8; ++nt) {
            v2f b = *(const v2f*)(&w1t[nt * 16 + row][k0 + klane]);
            acc[nt] = __builtin_amdgcn_wmma_f32_16x16x4_f32(
                false, a, false, b, (short)0, acc[nt], false, false);
        }
    }
    // K = 128..255: acceptor channels (same K index into w1t: concat layout)
#pragma unroll 2
    for (int k0 = CC; k0 < C2; k0 += 4) {
        v2f a = *(const v2f*)(arow + (k0 - CC) + klane);
#pragma unroll
        for (int nt = 0; nt < 8; ++nt) {
            v2f b = *(const v2f*)(&w1t[nt * 16 + row][k0 + klane]);
            acc[nt] = __builtin_amdgcn_wmma_f32_16x16x4_f32(
                false, a, false, b, (short)0, acc[nt], false, false);
        }
    }

    // Epilogue: pot[m] = sum_n relu(hid[m][n]) * W2[n] + b2.
    // Per lane: column n = nt*16 + row; reduce over the 16 lanes of the half.
    float part[8];
#pragma unroll
    for (int v = 0; v < 8; ++v) part[v] = 0.f;
#pragma unroll
    for (int nt = 0; nt < 8; ++nt) {
        float wv = W2[nt * 16 + row];
#pragma unroll
        for (int v = 0; v < 8; ++v) {
            float h = acc[nt][v];
            part[v] += (h > 0.f ? h : 0.f) * wv;
        }
    }
#pragma unroll
    for (int v = 0; v < 8; ++v) {
        float p = part[v];
        p += __shfl_xor(p, 1, 32);
        p += __shfl_xor(p, 2, 32);
        p += __shfl_xor(p, 4, 32);
        p += __shfl_xor(p, 8, 32);               // xor<16: stays within half
        part[v] = p;
    }
    if (row < 8) {                                // 16 writers: rows 0-7 / 8-15
        pot[g * JJ + j0 + row + half * 8] = part[row] + b2[0];
    }
}

// ---------------------------------------------------------------------------
// Kernel 2: per-gene transcript potentials, softmax over T+1, scatter-add of
// transcript probs onto donor/acceptor sites via LDS float atomics.
// One block per gene.
// ---------------------------------------------------------------------------
__global__ __launch_bounds__(256, 1)
void transcript_softmax_scatter(const float* __restrict__ pot,      // [G,J]
                                const float* __restrict__ refp,     // [1]
                                const int*   __restrict__ don_idx,  // [G,J]
                                const int*   __restrict__ acc_idx,  // [G,J]
                                const int*   __restrict__ tjunc,    // [G,T,K]
                                const float* __restrict__ tmask,    // [G,T,K]
                                float*       __restrict__ out)      // [G,S]
{
    __shared__ float spot[JJ];          // 8 KB: this gene's junction potentials
    __shared__ float sites[SS + 2];     // ~16 KB: scatter targets (incl. S,S+1)
    __shared__ float tpot[TT + 1];      // transcript potentials -> probs
    __shared__ float sred[2];           // max, 1/sum

    const int g   = blockIdx.x;
    const int tid = threadIdx.x;

    for (int i = tid; i < JJ; i += 256)     spot[i]  = pot[g * JJ + i];
    for (int i = tid; i < SS + 2; i += 256) sites[i] = 0.f;
    __syncthreads();

    if (tid < TT) {
        const int*   tj = tjunc + ((size_t)g * TT + tid) * KK;
        const float* tm = tmask + ((size_t)g * TT + tid) * KK;
        float s = 0.f;
#pragma unroll 4
        for (int k = 0; k < KK; ++k) s += spot[tj[k]] * tm[k];
        tpot[tid] = s;
    }
    if (tid == 0) tpot[TT] = refp[0];
    __syncthreads();

    // softmax over T+1 = 65 entries (tiny; serial reductions on thread 0)
    if (tid == 0) {
        float m = tpot[0];
        for (int t = 1; t <= TT; ++t) m = fmaxf(m, tpot[t]);
        sred[0] = m;
    }
    __syncthreads();
    if (tid <= TT) tpot[tid] = __expf(tpot[tid] - sred[0]);
    __syncthreads();
    if (tid == 0) {
        float s = 0.f;
        for (int t = 0; t <= TT; ++t) s += tpot[t];
        sred[1] = 1.f / s;
    }
    __syncthreads();
    if (tid <= TT) tpot[tid] *= sred[1];
    __syncthreads();

    // scatter: 2*T*K = 4096 ds_add_f32 atomics into the LDS site table
    for (int i = tid; i < TT * KK; i += 256) {
        int   t  = i >> 5;               // K == 32
        int   k  = i & 31;
        int   jj = tjunc[((size_t)g * TT + t) * KK + k];
        float p  = tpot[t] * tmask[((size_t)g * TT + t) * KK + k];
        atomicAdd(&sites[don_idx[g * JJ + jj]], p);
        atomicAdd(&sites[acc_idx[g * JJ + jj]], p);
    }
    __syncthreads();

    for (int i = tid; i < SS; i += 256)  // rows S, S+1 excluded per reference
        out[(size_t)g * SS + i] = sites[i];
}

// ---------------------------------------------------------------------------
extern "C" void kernel_launch(void* const* d_in, const int* in_sizes, int n_in,
                              void* d_out, int out_size, void* d_ws, size_t ws_size,
                              hipStream_t stream)
{
    const float* reps    = (const float*)d_in[0];   // [G,S,C]
    const float* gstart  = (const float*)d_in[1];   // [C]
    const float* gend    = (const float*)d_in[2];   // [C]
    const float* refp    = (const float*)d_in[3];   // [1]
    const float* W1      = (const float*)d_in[4];   // [2C,H]
    const float* b1      = (const float*)d_in[5];   // [H]
    const float* W2      = (const float*)d_in[6];   // [H,1]
    const float* b2      = (const float*)d_in[7];   // [1]
    const int*   don_idx = (const int*)d_in[8];     // [G,J]
    const int*   acc_idx = (const int*)d_in[9];     // [G,J]
    const int*   tjunc   = (const int*)d_in[10];    // [G,T,K]
    const float* tmask   = (const float*)d_in[11];  // [G,T,K]
    float* out = (float*)d_out;                     // [G,S]
    float* pot = (float*)d_ws;                      // [G,J] scratch (1 MB)

    (void)in_sizes; (void)n_in; (void)out_size; (void)ws_size;

    splice_junction_mlp<<<GG * (JJ / 128), 256, 0, stream>>>(
        reps, gstart, gend, W1, b1, W2, b2, don_idx, acc_idx, pot);

    transcript_softmax_scatter<<<GG, 256, 0, stream>>>(
        pot, refp, don_idx, acc_idx, tjunc, tmask, out);
}